// LocallyConnected2d_76476187672680
// MI455X (gfx1250) — compile-verified
//
#include <hip/hip_runtime.h>

// ---------------------------------------------------------------------------
// Locally-connected conv2d:  y[b,o,l] = sum_k x_unf[b,k,l] * w[o,k,l]
//   B=64, C_IN=96, H=W=32, C_OUT=192, K=3, PAD=1 -> L=1024, CKK=864
// Per-location GEMM [64x864]@[864x192], batched over 1024 locations.
// fp32 WMMA 16x16x4, tap-major contraction order:
//   K-axis of each WMMA = 4 input channels of one (kh,kw) tap.
//   -> kh/kw compile-time, no div/mod, uniform border predicates,
//      x row cached in registers across taps, float4 weight loads.
// ---------------------------------------------------------------------------

typedef __attribute__((ext_vector_type(2))) float v2f;
typedef __attribute__((ext_vector_type(8))) float v8f;

#define B_SZ   64
#define C_IN   96
#define H_IN   32
#define W_IN   32
#define C_OUT  192
#define H_OUT  32
#define W_OUT  32
#define L_SZ   (H_OUT * W_OUT)         // 1024
#define CKK    (C_IN * 9)              // 864
#define OWC    8                       // ow positions per workgroup (grid.z)

__global__ __launch_bounds__(128)
void lc2d_wmma_f32(const float* __restrict__ x,
                   const float* __restrict__ w,
                   float* __restrict__ y) {
  const int lane  = threadIdx.x & 31;
  const int wave  = threadIdx.x >> 5;     // 0..3 : M-tile (batch block)
  const int otile = blockIdx.x;           // 0..11 : N-tile (out-channel block)
  const int oh    = blockIdx.y;           // 0..31 : output row (l block)
  const int owc   = blockIdx.z * OWC;     // 0,8,16,24 : ow chunk (uniform)

  const int b0 = wave * 16;               // batch base of A tile
  const int o0 = otile * 16;              // channel base of B tile
  const int mn    = lane & 15;            // A row index == B col index
  const int kbase = (lane >> 4) * 2;      // this lane holds K = kstep+kbase+{0,1}

  // A rows come from x[b0+mn, c, ih, iw]
  const float* xrowbase = x + (size_t)(b0 + mn) * (C_IN * H_IN * W_IN);
  // B cols come from w[o0+mn, kk, oh*32 + ow]
  const float* wbase = w + (size_t)(o0 + mn) * CKK * L_SZ + (size_t)oh * W_OUT;

  const bool lE = (owc > 0);              // iw = owc-1 in range? (uniform)
  const bool rE = (owc + OWC < W_IN);     // iw = owc+8 in range? (uniform)

  v8f acc[OWC];
#pragma unroll
  for (int t = 0; t < OWC; ++t) acc[t] = (v8f)0.0f;

#pragma unroll
  for (int kh = 0; kh < 3; ++kh) {
    const int ih = oh + kh - 1;
    if ((unsigned)ih >= (unsigned)H_IN) continue;      // uniform: whole tap row is padding
    const float* xh = xrowbase + ih * W_IN;

    for (int c = 0; c < C_IN; c += 4) {
      const int cp = c + kbase;                        // this lane's channel pair
      const float* xr0 = xh + (size_t)cp * (H_IN * W_IN);
      const float* xr1 = xr0 + (H_IN * W_IN);

      // Register-cache x[cp, ih, owc-1 .. owc+8] and x[cp+1, ...]: 10 floats each.
      // a0[j] holds iw = owc-1+j;  A element for (kw,t) is a0[t+kw].
      float a0[OWC + 2], a1[OWC + 2];
      {
        const float4 p0 = *(const float4*)(xr0 + owc);
        const float4 p1 = *(const float4*)(xr0 + owc + 4);
        const float4 q0 = *(const float4*)(xr1 + owc);
        const float4 q1 = *(const float4*)(xr1 + owc + 4);
        a0[1] = p0.x; a0[2] = p0.y; a0[3] = p0.z; a0[4] = p0.w;
        a0[5] = p1.x; a0[6] = p1.y; a0[7] = p1.z; a0[8] = p1.w;
        a1[1] = q0.x; a1[2] = q0.y; a1[3] = q0.z; a1[4] = q0.w;
        a1[5] = q1.x; a1[6] = q1.y; a1[7] = q1.z; a1[8] = q1.w;
        a0[0]       = lE ? xr0[owc - 1]   : 0.0f;
        a1[0]       = lE ? xr1[owc - 1]   : 0.0f;
        a0[OWC + 1] = rE ? xr0[owc + OWC] : 0.0f;
        a1[OWC + 1] = rE ? xr1[owc + OWC] : 0.0f;
      }

#pragma unroll
      for (int kw = 0; kw < 3; ++kw) {
        // B rows for K = cp (bx) and K = cp+1 (by); 8 consecutive l's -> float4 x2
        const float* wb = wbase + ((size_t)cp * 9 + kh * 3 + kw) * L_SZ + owc;
        const float4 bx0 = *(const float4*)(wb);
        const float4 bx1 = *(const float4*)(wb + 4);
        const float4 by0 = *(const float4*)(wb + 9 * L_SZ);
        const float4 by1 = *(const float4*)(wb + 9 * L_SZ + 4);
        const float bx[OWC] = {bx0.x, bx0.y, bx0.z, bx0.w,
                               bx1.x, bx1.y, bx1.z, bx1.w};
        const float by[OWC] = {by0.x, by0.y, by0.z, by0.w,
                               by1.x, by1.y, by1.z, by1.w};

#pragma unroll
        for (int t = 0; t < OWC; ++t) {
          v2f a, b;
          a.x = a0[t + kw];   // A[m][cp]   = x[b0+m, cp,   ih, owc+t+kw-1]
          a.y = a1[t + kw];   // A[m][cp+1] = x[b0+m, cp+1, ih, owc+t+kw-1]
          b.x = bx[t];        // B[cp][n]   = w[o0+n, cp*9+tap,   l]
          b.y = by[t];        // B[cp+1][n] = w[o0+n, (cp+1)*9+tap, l]
          acc[t] = __builtin_amdgcn_wmma_f32_16x16x4_f32(
              /*neg_a=*/false, a, /*neg_b=*/false, b,
              /*c_mod=*/(short)0, acc[t], /*reuse_a=*/false, /*reuse_b=*/false);
        }
      }
    }
  }

  // C/D layout: VGPR r -> M = 8*(lane>>4)+r, N = lane&15
  const int mb = (lane >> 4) * 8;
#pragma unroll
  for (int t = 0; t < OWC; ++t) {
    const int l = oh * W_OUT + owc + t;
#pragma unroll
    for (int r = 0; r < 8; ++r) {
      y[((size_t)(b0 + mb + r) * C_OUT + (o0 + mn)) * L_SZ + l] = acc[t][r];
    }
  }
}

extern "C" void kernel_launch(void* const* d_in, const int* in_sizes, int n_in,
                              void* d_out, int out_size, void* d_ws, size_t ws_size,
                              hipStream_t stream) {
  const float* x = (const float*)d_in[0];   // [64, 96, 32, 32] f32
  const float* w = (const float*)d_in[1];   // [192, 864, 1024] f32
  float*       y = (float*)d_out;           // [64, 192, 32, 32] f32

  dim3 grid(C_OUT / 16, H_OUT, W_OUT / OWC);   // (12, 32, 4)
  dim3 block(128, 1, 1);                       // 4 waves = 4 M-tiles
  lc2d_wmma_f32<<<grid, block, 0, stream>>>(x, w, y);
}